// LocalAttention_24017457119616
// MI455X (gfx1250) — compile-verified
//
#include <hip/hip_runtime.h>

typedef float v2f __attribute__((ext_vector_type(2)));
typedef float v8f __attribute__((ext_vector_type(8)));
typedef unsigned int u32;
typedef unsigned long long u64;
typedef u32 v4u __attribute__((ext_vector_type(4)));
typedef u32 v8u __attribute__((ext_vector_type(8)));

#define FDIM 4096      // feature dim (= S*S) and rows (B*C = 64*64)
#define MT   64        // rows per workgroup
#define NT   128       // cols per workgroup (one h-group: 32 complete 2x2 blocks)
#define KC   16        // K chunk staged in LDS
#define LDSW 17        // padded stride for staging tiles (TDM pad: +1 DW per 16 DW)
#define EST  132       // padded stride for epilogue q/k/v planes
#define BUF_FLOATS (MT * LDSW + 3 * NT * LDSW)   // 7616 floats per staging buffer
#define EPI_FLOATS (3 * MT * EST)                // 25344 floats = ~99 KB dynamic LDS

// ---- Tensor Data Mover: DMA a (tile_rows x 16) fp32 tile with row stride 4096
// from global into LDS, inserting 1 DWORD pad per 16 DWORDs (-> stride 17). ----
__device__ __forceinline__ void tdm_load_2d(u32 lds_off_bytes, const float* g,
                                            u32 tile_rows)
{
    const u64 ga = (u64)(uintptr_t)g;
    v4u g0;
    g0[0] = 1u;                                   // count=1 (valid), no gather
    g0[1] = lds_off_bytes;                        // lds_addr
    g0[2] = (u32)ga;                              // global_addr[31:0]
    g0[3] = ((u32)(ga >> 32) & 0x01FFFFFFu)       // global_addr[56:32]
          | (2u << 30);                           // type = 2 ("image")
    v8u g1;
    g1[0] = (2u << 16)                            // data_size = 4 bytes
          | (1u << 20)                            // pad_enable
          | (3u << 22)                            // pad_interval: 16 DWORDs
          | (0u << 25);                           // pad_amount: 1 DWORD
    g1[1] = ((u32)FDIM & 0xFFFFu) << 16;          // tensor_dim0[15:0] (abar=0)
    g1[2] = ((u32)FDIM >> 16)                     // tensor_dim0[31:16]
          | (((u32)FDIM & 0xFFFFu) << 16);        // tensor_dim1[15:0]
    g1[3] = ((u32)FDIM >> 16)                     // tensor_dim1[31:16]
          | ((u32)KC << 16);                      // tile_dim0 = 16
    g1[4] = tile_rows;                            // tile_dim1 (tile_dim2 = 0)
    g1[5] = (u32)FDIM;                            // tensor_dim0_stride[31:0]
    g1[6] = 0u;                                   // stride hi / dim1_stride (2D)
    g1[7] = 0u;
    asm volatile("tensor_load_to_lds %0, %1" :: "s"(g0), "s"(g1) : "memory");
}

__global__ __launch_bounds__(256, 1)
void la_fused_kernel(const float* __restrict__ X,
                     const float* __restrict__ Wq,
                     const float* __restrict__ Wk,
                     const float* __restrict__ Wv,
                     float* __restrict__ Out)
{
    extern __shared__ float lds[];

    const int tid  = threadIdx.x;
    const int lane = tid & 31;          // wave32
    const int wave = tid >> 5;          // 8 waves
    const int half = lane >> 4;         // lane group 0/1
    const int l16  = lane & 15;
    const int wm   = wave & 3;          // M wave: 16 rows each (4x16 = 64)
    const int wn   = wave >> 2;         // N wave: 64 cols each (2x64 = 128)

    const int rowBase = blockIdx.x * MT;
    const int colBase = blockIdx.y * NT;

    // scalar (s_cbranch) leader test so only wave 0 *issues* TDM ops
    // (TDM ignores EXEC, so an exec-masked block would still issue 8x)
    const bool leader = (__builtin_amdgcn_readfirstlane(wave) == 0);
    const u32 ldsBase = (u32)(uintptr_t)(void*)lds;   // LDS byte offset

    // accumulators: 3 matrices x 4 N-tiles of 16x16 f32 (8 VGPRs each)
    v8f acc[3][4];
#pragma unroll
    for (int m = 0; m < 3; ++m)
#pragma unroll
        for (int t = 0; t < 4; ++t)
#pragma unroll
            for (int i = 0; i < 8; ++i)
                acc[m][t][i] = 0.0f;

    const int NCHUNK = FDIM / KC;   // 256

    // issue chunk 0 into buffer 0
    if (leader) {
        const u32 b = ldsBase;
        tdm_load_2d(b, X + (size_t)rowBase * FDIM, MT);
        tdm_load_2d(b + (u32)(MT * LDSW) * 4u,
                    Wq + (size_t)colBase * FDIM, NT);
        tdm_load_2d(b + (u32)(MT * LDSW + NT * LDSW) * 4u,
                    Wk + (size_t)colBase * FDIM, NT);
        tdm_load_2d(b + (u32)(MT * LDSW + 2 * NT * LDSW) * 4u,
                    Wv + (size_t)colBase * FDIM, NT);
    }

    for (int i = 0; i < NCHUNK; ++i) {
        const int cur = i & 1;
        if (leader) {
            if (i + 1 < NCHUNK) {
                // prefetch chunk i+1 into the other buffer (DMA overlaps WMMAs)
                const int k1 = (i + 1) * KC;
                const u32 b = ldsBase + (u32)((1 - cur) * BUF_FLOATS) * 4u;
                tdm_load_2d(b, X + (size_t)rowBase * FDIM + k1, MT);
                tdm_load_2d(b + (u32)(MT * LDSW) * 4u,
                            Wq + (size_t)colBase * FDIM + k1, NT);
                tdm_load_2d(b + (u32)(MT * LDSW + NT * LDSW) * 4u,
                            Wk + (size_t)colBase * FDIM + k1, NT);
                tdm_load_2d(b + (u32)(MT * LDSW + 2 * NT * LDSW) * 4u,
                            Wv + (size_t)colBase * FDIM + k1, NT);
                __builtin_amdgcn_s_wait_tensorcnt(4);   // chunk i landed
            } else {
                __builtin_amdgcn_s_wait_tensorcnt(0);   // last chunk landed
            }
        }
        __syncthreads();   // chunk i visible to all waves

        const float* Xl = lds + cur * BUF_FLOATS;          // 64 x KC, stride 17
        const float* Wl = Xl + MT * LDSW;                  // 3 x (128 x KC)

#pragma unroll
        for (int kk = 0; kk < KC; kk += 4) {
            // A 16x4 f32: lane L -> M = L%16, VGPR v holds K = 2*(L/16) + v
            v2f a;
            {
                const float* s = Xl + (16 * wm + l16) * LDSW + kk + half * 2;
                a[0] = s[0];
                a[1] = s[1];
            }
            // ---- load ALL 12 B fragments into distinct registers first so the
            // ds_loads batch up and WMMAs are not serialized on dscnt 0 ----
            v2f bfrag[12];
#pragma unroll
            for (int m = 0; m < 3; ++m) {
                const float* wl = Wl + m * (NT * LDSW);
#pragma unroll
                for (int t = 0; t < 4; ++t) {
                    // B 4x16 f32: lane L -> N = L%16, VGPR v holds K = 2v + (L/16)
                    const int n = 64 * wn + 16 * t + l16;
                    bfrag[m * 4 + t][0] = wl[n * LDSW + kk + half];
                    bfrag[m * 4 + t][1] = wl[n * LDSW + kk + half + 2];
                }
            }
#pragma unroll
            for (int m = 0; m < 3; ++m)
#pragma unroll
                for (int t = 0; t < 4; ++t)
                    acc[m][t] = __builtin_amdgcn_wmma_f32_16x16x4_f32(
                        false, a, false, bfrag[m * 4 + t], (short)0,
                        acc[m][t], false, false);
        }
        __syncthreads();   // all reads of buf[cur] done before it is re-filled
    }

    // ---- spill accumulators to LDS q/k/v planes (64 x 128, stride EST) ----
    // C/D layout: VGPR i, lanes 0-15: M=i, lanes 16-31: M=8+i; N = L%16
#pragma unroll
    for (int m = 0; m < 3; ++m) {
        float* pl = lds + m * (MT * EST);
#pragma unroll
        for (int t = 0; t < 4; ++t) {
            const int col = 64 * wn + 16 * t + l16;
#pragma unroll
            for (int i = 0; i < 8; ++i) {
                const int row = 16 * wm + 8 * half + i;
                pl[row * EST + col] = acc[m][t][i];
            }
        }
    }
    __syncthreads();

    // ---- 2x2 local-attention epilogue ----
    // block (h=blockIdx.y, w): features {2w, 2w+1, 64+2w, 65+2w} within this tile,
    // output cols 4w+g within this tile (contiguous).
    for (int task = tid; task < MT * 32; task += 256) {
        const int r = task >> 5;
        const int w = task & 31;
        const float* ql = lds + r * EST;
        const float* kl = lds + MT * EST + r * EST;
        const float* vl = lds + 2 * MT * EST + r * EST;
        const int c0 = 2 * w, c1 = 2 * w + 1, c2 = 64 + 2 * w, c3 = 65 + 2 * w;
        const float q[4]  = {ql[c0], ql[c1], ql[c2], ql[c3]};
        const float kv[4] = {kl[c0], kl[c1], kl[c2], kl[c3]};
        const float vv[4] = {vl[c0], vl[c1], vl[c2], vl[c3]};
        float o[4];
#pragma unroll
        for (int g = 0; g < 4; ++g) {
            const float s0 = q[g] * kv[0], s1 = q[g] * kv[1];
            const float s2 = q[g] * kv[2], s3 = q[g] * kv[3];
            const float mx = fmaxf(fmaxf(s0, s1), fmaxf(s2, s3));
            const float e0 = __expf(s0 - mx), e1 = __expf(s1 - mx);
            const float e2 = __expf(s2 - mx), e3 = __expf(s3 - mx);
            const float inv = 1.0f / (e0 + e1 + e2 + e3);
            o[g] = (e0 * vv[0] + e1 * vv[1] + e2 * vv[2] + e3 * vv[3]) * inv;
        }
        float4 ov = {o[0], o[1], o[2], o[3]};
        *(float4*)(Out + (size_t)(rowBase + r) * FDIM + colBase + 4 * w) = ov;
    }
}

extern "C" void kernel_launch(void* const* d_in, const int* in_sizes, int n_in,
                              void* d_out, int out_size, void* d_ws, size_t ws_size,
                              hipStream_t stream)
{
    const float* x  = (const float*)d_in[0];
    const float* Wq = (const float*)d_in[1];
    const float* Wk = (const float*)d_in[2];
    const float* Wv = (const float*)d_in[3];
    float* out = (float*)d_out;

    dim3 grid(FDIM / MT, FDIM / NT);   // (64, 32): x = row tile, y = column group h
    la_fused_kernel<<<grid, dim3(256), EPI_FLOATS * sizeof(float), stream>>>(
        x, Wq, Wk, Wv, out);
}